// FinePredictor_48060684042813
// MI455X (gfx1250) — compile-verified
//
#include <hip/hip_runtime.h>

// Problem constants (from reference: B=64, S=2048, D=512, N_FINE=16, MAX_SPANS=512)
#define BSZ   64
#define SEQ   2048
#define DIM   512
#define NF    16
#define MS    512

typedef __attribute__((ext_vector_type(2))) float v2f;
typedef __attribute__((ext_vector_type(8))) float v8f;

// ---------------------------------------------------------------------------
// Kernel 0: zero the output (harness poisons d_out with 0xAA).
// out has 64*512*16 = 524288 floats = 131072 float4. 512 blocks x 256 threads.
// ---------------------------------------------------------------------------
__global__ void __launch_bounds__(256) zero_out_kernel(float4* __restrict__ out) {
  size_t i = (size_t)blockIdx.x * blockDim.x + threadIdx.x;
  out[i] = make_float4(0.f, 0.f, 0.f, 0.f);
}

// ---------------------------------------------------------------------------
// Kernel 1: BIO segment ids. One block per batch row (64 blocks, 256 threads).
// Thread t owns tokens [t*8, t*8+8). Local B-tag counts -> block prefix scan
// -> per-token inclusive span id; invalid tokens get -1.
// ---------------------------------------------------------------------------
__global__ void __launch_bounds__(256) seg_id_kernel(const int* __restrict__ labels,
                                                     const int* __restrict__ pBtag,
                                                     const int* __restrict__ pItag,
                                                     const int* __restrict__ pMaxSpans,
                                                     int* __restrict__ seg_out) {
  __shared__ int cnts[256];
  __shared__ int pref[256];
  const int b    = blockIdx.x;
  const int t    = threadIdx.x;
  const int Btag = *pBtag;
  const int Itag = *pItag;
  const int maxs = *pMaxSpans;

  const int* row = labels + (size_t)b * SEQ;
  const int base = t * (SEQ / 256);   // 8 tokens per thread

  int lab[SEQ / 256];
  int c = 0;
#pragma unroll
  for (int i = 0; i < SEQ / 256; ++i) {
    lab[i] = row[base + i];
    c += (lab[i] == Btag) ? 1 : 0;
  }
  cnts[t] = c;
  __syncthreads();

  if (t == 0) {
    int run = 0;
    for (int i = 0; i < 256; ++i) { pref[i] = run; run += cnts[i]; }
  }
  __syncthreads();

  int running = pref[t];  // exclusive count of B tags before my first token
  int* orow = seg_out + (size_t)b * SEQ;
#pragma unroll
  for (int i = 0; i < SEQ / 256; ++i) {
    const bool isB = (lab[i] == Btag);
    const bool isI = (lab[i] == Itag);
    running += isB ? 1 : 0;
    const int seg = running - 1;     // cumsum(isB) - 1, inclusive
    const bool valid = (isB || isI) && (seg >= 0) && (seg < maxs);
    orow[base + i] = valid ? seg : -1;
  }
}

// ---------------------------------------------------------------------------
// Kernel 2: project + scatter. Grid = (B*S)/128 = 1024 blocks, 256 threads
// (8 waves). Each wave owns 16 contiguous tokens of one batch row and runs a
// 16(tokens) x 512(K) x 16(slots) GEMM via v_wmma_f32_16x16x4_f32, then
// atomically accumulates each token row into out[b, seg(token), :].
//
// A frag (16x4 f32): lane l -> row M = l&15; VGPR0/1 hold K = 2*(l>>4)+{0,1}.
// B frag (4x16 f32): lane l -> col N = l&15; VGPR0/1 hold K = 2*(l>>4)+{0,1}.
// C (16x16 f32, 8 VGPRs): lane l, VGPR r -> M = (l>>4)*8 + r, N = l&15.
// ---------------------------------------------------------------------------
__global__ void __launch_bounds__(256)
fine_predict_kernel(const float* __restrict__ hidden,
                    const float* __restrict__ slot,      // [DIM, NF] row-major
                    const int*   __restrict__ seg_ids,   // [B*S], -1 = drop
                    float*       __restrict__ out) {     // [B, MS, NF]
  __shared__ float sB[DIM * NF];  // 32 KB, layout [k][f]

  const int tid = threadIdx.x;

  // Stage slot_embs into LDS with coalesced b128 loads.
  {
    const float4* s4 = (const float4*)slot;
    float4* l4 = (float4*)sB;
#pragma unroll
    for (int i = tid; i < (DIM * NF) / 4; i += 256) l4[i] = s4[i];
  }
  __syncthreads();

  const int wave   = tid >> 5;
  const int lane   = tid & 31;
  const int n      = lane & 15;        // slot column
  const int khalf  = lane >> 4;        // 0 or 1: selects K pair within step
  const int token0 = blockIdx.x * 128 + wave * 16;

  // A row pointer for this lane (row m = lane&15, K offset 2*khalf)
  const float* arow = hidden + (size_t)(token0 + n) * DIM + 2 * khalf;

  v8f c = {};
#pragma unroll 8
  for (int k = 0; k < DIM; k += 4) {
    v2f a = *(const v2f*)(arow + k);
    const int kb = k + 2 * khalf;
    v2f bf;
    bf.x = sB[kb * NF + n];
    bf.y = sB[(kb + 1) * NF + n];
    c = __builtin_amdgcn_wmma_f32_16x16x4_f32(
        /*neg_a=*/false, a, /*neg_b=*/false, bf,
        /*c_mod=*/(short)0, c, /*reuse_a=*/false, /*reuse_b=*/false);
  }

  // Scatter-accumulate: lane holds column n of rows m = khalf*8 + r.
  // Hoist the 8 contiguous (32B-aligned) segment ids into two b128 loads so
  // the epilogue is [1 wait] -> 8 predicated atomics, not 8 load/wait chains.
  const int batch = token0 / SEQ;
  const int mbase = khalf * 8;
  const int4 s0 = *(const int4*)(seg_ids + token0 + mbase);
  const int4 s1 = *(const int4*)(seg_ids + token0 + mbase + 4);
  int segs[8] = {s0.x, s0.y, s0.z, s0.w, s1.x, s1.y, s1.z, s1.w};

  float* obase = out + ((size_t)batch * MS) * NF + n;
#pragma unroll
  for (int r = 0; r < 8; ++r) {
    const int seg = segs[r];
    if (seg >= 0) {
      atomicAdd(obase + (size_t)seg * NF, c[r]);
    }
  }
}

// ---------------------------------------------------------------------------
// Launch
// ---------------------------------------------------------------------------
extern "C" void kernel_launch(void* const* d_in, const int* in_sizes, int n_in,
                              void* d_out, int out_size, void* d_ws, size_t ws_size,
                              hipStream_t stream) {
  const float* hidden = (const float*)d_in[0];  // [B, S, D] f32
  const float* slot   = (const float*)d_in[1];  // [D, NF]   f32
  const int*   labels = (const int*)d_in[2];    // [B, S]    i32
  const int*   pBtag  = (const int*)d_in[3];    // scalar
  const int*   pItag  = (const int*)d_in[4];    // scalar
  const int*   pMaxS  = (const int*)d_in[5];    // scalar

  float* out    = (float*)d_out;                // [B, MS, NF] f32
  int*   seg_ws = (int*)d_ws;                   // [B*S] i32 = 512 KB scratch

  // 1) zero output: 524288 floats = 131072 float4 -> 512 blocks x 256
  zero_out_kernel<<<512, 256, 0, stream>>>((float4*)out);

  // 2) segment ids: one block per batch row
  seg_id_kernel<<<BSZ, 256, 0, stream>>>(labels, pBtag, pItag, pMaxS, seg_ws);

  // 3) WMMA projection + atomic scatter: (B*S)/128 = 1024 blocks
  fine_predict_kernel<<<(BSZ * SEQ) / 128, 256, 0, stream>>>(hidden, slot, seg_ws, out);
}